// SpinSphericalResidualBlock_31310311588285
// MI455X (gfx1250) — compile-verified
//
#include <hip/hip_runtime.h>
#include <hip/hip_bf16.h>
#include <math.h>

// ---------------------------------------------------------------------------
// SpinSphericalResidualBlock on MI455X (gfx1250, wave32, WMMA).
//
// Every contraction (DFT-as-matmul, Legendre transform, channel mix, residual
// mix) is expressed as batched f32 GEMMs on V_WMMA_F32_16X16X4_F32. The
// problem is HBM-bound (~1.5 GB moved, ~35 GFLOP), so fp32 WMMA is the right
// precision: keeps reference accuracy and the matrix ceiling is irrelevant.
// MBINS gather/scatter is folded into precomputed DFT matrices; phase_collapse
// is fused into the inverse-DFT kernel so spatial fields stay real.
// ---------------------------------------------------------------------------

typedef __attribute__((ext_vector_type(2))) float v2f;
typedef __attribute__((ext_vector_type(8))) float v8f;

#define B_ 8
#define N_ 128
#define S_ 2
#define C_ 32
#define L_ 64
#define M_ 127
#define ROWS_ (L_*B_*M_*S_)        // 130048 rows of (s,c)-vectors in coeff layout
#define NCPLX_ ((size_t)ROWS_*C_)  // 4,161,536 complex coefficients

// Fragment lane mapping for 16x16x4 f32 WMMA (wave32), per ISA 7.12.2:
//   A (16x4):  row = lane&15,  k pair = (lane>>4)*2 .. +1
//   B (4x16):  col = lane&15,  k pair = (lane>>4)*2 .. +1
//   C/D:       col = lane&15,  VGPR i -> row i + 8*(lane>>4)
__device__ __forceinline__ v8f wmma4(v2f a, v2f b, v8f c) {
  return __builtin_amdgcn_wmma_f32_16x16x4_f32(false, a, false, b, (short)0, c,
                                               false, false);
}

// ---------------------------------------------------------------------------
// K0: build DFT matrices with MBINS folded in.
//   Wf[m][y] = exp(-2pi*i*mbin(m)*y/N)/N   (row 127 = 0)    [m-major]
//   Wb[y][m] = exp(+2pi*i*mbin(m)*y/N)     (col 127 = 0)    [y-major]
__global__ void k_init_dft(float* __restrict__ wfr, float* __restrict__ wfi,
                           float* __restrict__ wbr, float* __restrict__ wbi) {
  int m = blockIdx.x;   // 0..127
  int y = threadIdx.x;  // 0..127
  float fr = 0.f, fi = 0.f, br = 0.f, bi = 0.f;
  if (m < 127) {
    int mb = (m - 63) & 127;  // MBINS
    float ang = 6.283185307179586f * (float)((mb * y) & 127) * (1.0f / 128.0f);
    float cs = cosf(ang), sn = sinf(ang);
    fr = cs * (1.0f / 128.0f);
    fi = -sn * (1.0f / 128.0f);
    br = cs;
    bi = sn;
  }
  wfr[m * 128 + y] = fr;
  wfi[m * 128 + y] = fi;
  wbr[y * 128 + m] = br;
  wbi[y * 128 + m] = bi;
}

// ---------------------------------------------------------------------------
// K1: forward DFT along axis 2 (real input), MBINS gathered.
//   fm[s][m][t][b][c] = sum_y Wf[m][y] * x[b][t][y][s][c]
// One block per (b,t,s): out tile 128(m) x 32(c), K=128(y).
__global__ __launch_bounds__(256)
void k_dft_fwd(const float* __restrict__ xin, const float* __restrict__ wfr,
               const float* __restrict__ wfi, float* __restrict__ fm) {
  extern __shared__ float sm[];
  float* sWr = sm;              // 16384
  float* sWi = sm + 16384;      // 16384
  float* sX  = sm + 32768;      // 4096  (128y x 32c real)
  int bid = blockIdx.x;
  int s = bid & 1, t = (bid >> 1) & 127, b = bid >> 8;
  int tid = threadIdx.x;
  for (int i = tid; i < 16384; i += 256) { sWr[i] = wfr[i]; sWi[i] = wfi[i]; }
  const float* xb = xin + (((size_t)(b * 128 + t) * 128) * 2 + s) * 32;
  for (int i = tid; i < 4096; i += 256) {
    int y = i >> 5, c = i & 31;
    sX[i] = xb[(size_t)y * 64 + c];
  }
  __syncthreads();
  int lane = tid & 31, wid = tid >> 5;
  int rowA = lane & 15, ks0 = (lane >> 4) * 2;
  int mt = wid;  // 8 m-tiles
  for (int ct = 0; ct < 2; ++ct) {
    int col = ct * 16 + (lane & 15);
    v8f accr = {}, acci = {};
    for (int k0 = 0; k0 < 128; k0 += 4) {
      int ka = k0 + ks0;
      v2f ar, ai, bx;
      int arow = (mt * 16 + rowA) * 128 + ka;
      ar.x = sWr[arow]; ar.y = sWr[arow + 1];
      ai.x = sWi[arow]; ai.y = sWi[arow + 1];
      bx.x = sX[ka * 32 + col]; bx.y = sX[(ka + 1) * 32 + col];
      accr = wmma4(ar, bx, accr);
      acci = wmma4(ai, bx, acci);
    }
    int mbase = mt * 16 + 8 * (lane >> 4);
    for (int i = 0; i < 8; ++i) {
      int m = mbase + i;
      if (m < 127) {
        size_t idx = ((((size_t)s * 127 + m) * 128 + t) * 8 + b) * 32 + col;
        fm[idx * 2] = accr[i];
        fm[idx * 2 + 1] = acci[i];
      }
    }
  }
}

// ---------------------------------------------------------------------------
// K2: forward Legendre. One block per (s,m):
//   out[l][b][m][s][c] = sum_t leg[s][l][m][t] * fm[s][m][t][b][c]
// A = leg tile (64x128, real) staged in LDS; X read from global (L2-hot).
__global__ __launch_bounds__(256)
void k_leg_fwd(const float* __restrict__ leg, const float* __restrict__ fm,
               float* __restrict__ outc) {
  extern __shared__ float sA[];  // 64*128 = 8192 floats
  int bid = blockIdx.x;
  int s = bid / 127, m = bid % 127;
  int tid = threadIdx.x;
  for (int i = tid; i < 8192; i += 256) {
    int l = i >> 7, t = i & 127;
    sA[i] = leg[(((size_t)s * 64 + l) * 127 + m) * 128 + t];
  }
  __syncthreads();
  const float* Xb = fm + (((size_t)s * 127 + m) * 128) * 256 * 2;  // [t][(b,c)] cplx
  int lane = tid & 31, wid = tid >> 5;
  int rowA = lane & 15, ks0 = (lane >> 4) * 2;
  for (int tile = wid; tile < 64; tile += 8) {  // 4 l-tiles x 16 col-tiles
    int lt = tile >> 4, ct = tile & 15;
    int col = ct * 16 + (lane & 15);
    v8f accr = {}, acci = {};
    for (int k0 = 0; k0 < 128; k0 += 4) {
      int ka = k0 + ks0;
      v2f a, br, bi;
      int arow = (lt * 16 + rowA) * 128 + ka;
      a.x = sA[arow]; a.y = sA[arow + 1];
      const float* p0 = Xb + ((size_t)ka * 256 + col) * 2;
      const float* p1 = Xb + ((size_t)(ka + 1) * 256 + col) * 2;
      br.x = p0[0]; bi.x = p0[1];
      br.y = p1[0]; bi.y = p1[1];
      accr = wmma4(a, br, accr);
      acci = wmma4(a, bi, acci);
    }
    int b = col >> 5, c = col & 31;
    int lbase = lt * 16 + 8 * (lane >> 4);
    for (int i = 0; i < 8; ++i) {
      int l = lbase + i;
      size_t idx = ((((size_t)l * 8 + b) * 127 + m) * 2 + s) * 32 + c;
      outc[idx * 2] = accr[i];
      outc[idx * 2 + 1] = acci[i];
    }
  }
}

// ---------------------------------------------------------------------------
// K3: complex channel mix  h[b,l,m,t,d] = sum_{s,c} X[b,l,m,s,c]*W[l,s,t,c,d]
// Per l: 1016 rows x 64 cols, K=64, complex x complex (4 real WMMAs / k-step).
// Grid = L*8 blocks (128 rows each); W tile (64x64 cplx) staged in LDS.
__global__ __launch_bounds__(256)
void k_chanmix(const float* __restrict__ X, const float* __restrict__ wre,
               const float* __restrict__ wim, float* __restrict__ Y) {
  extern __shared__ float sm[];
  float* sWr = sm;          // 4096
  float* sWi = sm + 4096;   // 4096
  int bid = blockIdx.x;
  int l = bid >> 3, rg = bid & 7;
  int tid = threadIdx.x;
  for (int i = tid; i < 4096; i += 256) {
    int k = i >> 6, col = i & 63;
    int s = k >> 5, c = k & 31, t = col >> 5, d = col & 31;
    size_t widx = (((size_t)l * 2 + s) * 2 + t) * 1024 + c * 32 + d;
    sWr[i] = wre[widx];
    sWi[i] = wim[widx];
  }
  __syncthreads();
  const float* Xl = X + (size_t)l * 1016 * 64 * 2;
  float* Yl = Y + (size_t)l * 1016 * 64 * 2;
  int lane = tid & 31, wid = tid >> 5;
  int ks0 = (lane >> 4) * 2;
  for (int tile = wid; tile < 32; tile += 8) {  // 8 row-tiles x 4 col-tiles
    int rt = tile >> 2, ct = tile & 3;
    int row0 = rg * 128 + rt * 16;
    int row = row0 + (lane & 15);
    int rr = (row < 1016) ? row : 0;
    int col = ct * 16 + (lane & 15);
    v8f accr = {}, acci = {};
    for (int k0 = 0; k0 < 64; k0 += 4) {
      int ka = k0 + ks0;
      const float* px = Xl + ((size_t)rr * 64 + ka) * 2;
      v2f axr, axi, axin, bwr, bwi;
      axr.x = px[0]; axi.x = px[1];
      axr.y = px[2]; axi.y = px[3];
      axin.x = -axi.x; axin.y = -axi.y;  // F32 WMMA NEG only covers C -> VALU negate
      bwr.x = sWr[ka * 64 + col]; bwr.y = sWr[(ka + 1) * 64 + col];
      bwi.x = sWi[ka * 64 + col]; bwi.y = sWi[(ka + 1) * 64 + col];
      accr = wmma4(axr, bwr, accr);
      accr = wmma4(axin, bwi, accr);
      acci = wmma4(axr, bwi, acci);
      acci = wmma4(axi, bwr, acci);
    }
    int rbase = row0 + 8 * (lane >> 4);
    for (int i = 0; i < 8; ++i) {
      int r = rbase + i;
      if (r < 1016) {
        size_t idx = (size_t)r * 64 + col;
        Yl[idx * 2] = accr[i];
        Yl[idx * 2 + 1] = acci[i];
      }
    }
  }
}

// ---------------------------------------------------------------------------
// K4: residual mix res[...,s,d] = sum_c X[...,s,c]*wd[c,d] + bd[d]
// 130048 rows x 32 cols, K=32, real W. Grid=1016, 8 row-tiles/block.
__global__ __launch_bounds__(256)
void k_resmix(const float* __restrict__ X, const float* __restrict__ wd,
              const float* __restrict__ bd, float* __restrict__ Y) {
  __shared__ float sW[1024];
  __shared__ float sB[32];
  int tid = threadIdx.x;
  for (int i = tid; i < 1024; i += 256) sW[i] = wd[i];
  if (tid < 32) sB[tid] = bd[tid];
  __syncthreads();
  int lane = tid & 31, wid = tid >> 5;
  int ks0 = (lane >> 4) * 2;
  int rt = blockIdx.x * 8 + wid;  // row tile 0..8127
  int row = rt * 16 + (lane & 15);
  for (int ct = 0; ct < 2; ++ct) {
    int col = ct * 16 + (lane & 15);
    v8f accr = {}, acci = {};
    for (int k0 = 0; k0 < 32; k0 += 4) {
      int ka = k0 + ks0;
      const float* px = X + ((size_t)row * 32 + ka) * 2;
      v2f ar, ai, bw;
      ar.x = px[0]; ai.x = px[1];
      ar.y = px[2]; ai.y = px[3];
      bw.x = sW[ka * 32 + col]; bw.y = sW[(ka + 1) * 32 + col];
      accr = wmma4(ar, bw, accr);
      acci = wmma4(ai, bw, acci);
    }
    int rbase = rt * 16 + 8 * (lane >> 4);
    for (int i = 0; i < 8; ++i) {
      size_t idx = (size_t)(rbase + i) * 32 + col;
      Y[idx * 2] = accr[i] + sB[col];
      Y[idx * 2 + 1] = acci[i];
    }
  }
}

// ---------------------------------------------------------------------------
// K5: backward Legendre. One block per (s,m):
//   fm[b][t][s][m][c] = sum_l leg_bwd[s][l][m][t] * H[l][b][m][s][c]
__global__ __launch_bounds__(256)
void k_leg_bwd(const float* __restrict__ leg, const float* __restrict__ H,
               float* __restrict__ fm) {
  extern __shared__ float sA[];  // [l][t] 8192 floats; read transposed
  int bid = blockIdx.x;
  int s = bid / 127, m = bid % 127;
  int tid = threadIdx.x;
  for (int i = tid; i < 8192; i += 256) {
    int l = i >> 7, t = i & 127;
    sA[i] = leg[(((size_t)s * 64 + l) * 127 + m) * 128 + t];
  }
  __syncthreads();
  int lane = tid & 31, wid = tid >> 5;
  int ks0 = (lane >> 4) * 2;
  for (int tile = wid; tile < 128; tile += 8) {  // 8 t-tiles x 16 col-tiles
    int rt = tile >> 4, ct = tile & 15;
    int col = ct * 16 + (lane & 15);
    int b = col >> 5, c = col & 31;
    int trow = rt * 16 + (lane & 15);
    v8f accr = {}, acci = {};
    for (int k0 = 0; k0 < 64; k0 += 4) {
      int ka = k0 + ks0;
      v2f a, br, bi;
      a.x = sA[ka * 128 + trow];        // A[t][l] = legT
      a.y = sA[(ka + 1) * 128 + trow];
      size_t h0 = ((((size_t)ka * 8 + b) * 127 + m) * 2 + s) * 32 + c;
      size_t h1 = ((((size_t)(ka + 1) * 8 + b) * 127 + m) * 2 + s) * 32 + c;
      br.x = H[h0 * 2]; bi.x = H[h0 * 2 + 1];
      br.y = H[h1 * 2]; bi.y = H[h1 * 2 + 1];
      accr = wmma4(a, br, accr);
      acci = wmma4(a, bi, acci);
    }
    int tbase = rt * 16 + 8 * (lane >> 4);
    for (int i = 0; i < 8; ++i) {
      int t = tbase + i;
      size_t idx = ((((size_t)b * 128 + t) * 2 + s) * 127 + m) * 32 + c;
      fm[idx * 2] = accr[i];
      fm[idx * 2 + 1] = acci[i];
    }
  }
}

// ---------------------------------------------------------------------------
// K6: inverse DFT (MBINS scatter folded) + fused phase_collapse.
//   f[b][t][y][s][c] = sum_m Wb[y][m] * fm[b][t][s][m][c]  -> collapse
__global__ __launch_bounds__(256)
void k_dft_bwd(const float* __restrict__ fm, const float* __restrict__ wbr,
               const float* __restrict__ wbi, float* __restrict__ outR,
               float* __restrict__ outC, int finalOut) {
  extern __shared__ float sm[];
  float* sWr = sm;                 // 16384
  float* sWi = sm + 16384;         // 16384
  float* sXr = sm + 32768;         // 4096 (128m x 32c, row 127 zero-padded)
  float* sXi = sm + 32768 + 4096;  // 4096
  int bid = blockIdx.x;
  int s = bid & 1, t = (bid >> 1) & 127, b = bid >> 8;
  int tid = threadIdx.x;
  for (int i = tid; i < 16384; i += 256) { sWr[i] = wbr[i]; sWi[i] = wbi[i]; }
  const float* Xb = fm + ((((size_t)b * 128 + t) * 2 + s) * 127) * 32 * 2;
  for (int i = tid; i < 4096; i += 256) {
    int mrow = i >> 5, c = i & 31;
    float re = 0.f, im = 0.f;
    if (mrow < 127) {
      re = Xb[((size_t)mrow * 32 + c) * 2];
      im = Xb[((size_t)mrow * 32 + c) * 2 + 1];
    }
    sXr[i] = re;
    sXi[i] = im;
  }
  __syncthreads();
  int lane = tid & 31, wid = tid >> 5;
  int rowA = lane & 15, ks0 = (lane >> 4) * 2;
  int yt = wid;
  for (int ct = 0; ct < 2; ++ct) {
    int col = ct * 16 + (lane & 15);
    v8f accr = {}, acci = {};
    for (int k0 = 0; k0 < 128; k0 += 4) {
      int ka = k0 + ks0;
      v2f ar, ai, ain, br, bi;
      int arow = (yt * 16 + rowA) * 128 + ka;
      ar.x = sWr[arow]; ar.y = sWr[arow + 1];
      ai.x = sWi[arow]; ai.y = sWi[arow + 1];
      ain.x = -ai.x; ain.y = -ai.y;
      br.x = sXr[ka * 32 + col]; br.y = sXr[(ka + 1) * 32 + col];
      bi.x = sXi[ka * 32 + col]; bi.y = sXi[(ka + 1) * 32 + col];
      accr = wmma4(ar, br, accr);
      accr = wmma4(ain, bi, accr);
      acci = wmma4(ar, bi, acci);
      acci = wmma4(ai, br, acci);
    }
    int ybase = yt * 16 + 8 * (lane >> 4);
    for (int i = 0; i < 8; ++i) {
      int y = ybase + i;
      float re = accr[i], im = acci[i];
      float val = (s == 0) ? fmaxf(re, 0.f) : sqrtf(re * re + im * im + 1e-12f);
      size_t idx = ((((size_t)b * 128 + t) * 128 + y) * 2 + s) * 32 + col;
      if (finalOut) {
        outC[idx * 2] = val;       // complex64 output, imag = 0
        outC[idx * 2 + 1] = 0.f;
      } else {
        outR[idx] = val;
      }
    }
  }
}

// ---------------------------------------------------------------------------
// Spectral BN pieces (c layout: rows r = (((l*8+b)*127+m)*2+s), 32 cplx each)
__global__ void k_bn_mean(float* __restrict__ c) {  // subtract batch mean @ l=0,m=63,s=0
  int tid = threadIdx.x;
  if (tid < 32) {
    float mr = 0.f, mi = 0.f;
    for (int b = 0; b < 8; ++b) {
      size_t idx = ((((size_t)b) * 127 + 63) * 2 + 0) * 32 + tid;
      mr += c[idx * 2];
      mi += c[idx * 2 + 1];
    }
    mr *= 0.125f; mi *= 0.125f;
    for (int b = 0; b < 8; ++b) {
      size_t idx = ((((size_t)b) * 127 + 63) * 2 + 0) * 32 + tid;
      c[idx * 2] -= mr;
      c[idx * 2 + 1] -= mi;
    }
  }
}

__global__ void k_zero_var(float* __restrict__ var) {
  if (threadIdx.x < 64) var[threadIdx.x] = 0.f;
}

__global__ __launch_bounds__(256)
void k_bn_var(const float* __restrict__ c, float* __restrict__ var) {
  __shared__ float red[512];
  int tid = threadIdx.x;
  int cc = tid & 31, rl = tid >> 5;
  float a0 = 0.f, a1 = 0.f;
  size_t r0 = (size_t)blockIdx.x * 128;
  for (int p = 0; p < 16; ++p) {
    size_t r = r0 + (size_t)p * 8 + rl;
    const float* pz = c + (r * 32 + cc) * 2;
    float v = pz[0] * pz[0] + pz[1] * pz[1];
    if ((r & 1) == 0) a0 += v; else a1 += v;
  }
  red[tid * 2] = a0;
  red[tid * 2 + 1] = a1;
  __syncthreads();
  if (tid < 64) {
    int s = tid >> 5, ccc = tid & 31;
    float acc = 0.f;
    for (int rr = 0; rr < 8; ++rr) acc += red[(rr * 32 + ccc) * 2 + s];
    atomicAdd(&var[s * 32 + ccc], acc);
  }
}

__global__ void k_bn_factor(const float* __restrict__ var,
                            const float* __restrict__ scale,
                            float* __restrict__ factor) {
  int i = threadIdx.x;
  if (i < 64) factor[i] = scale[i] * rsqrtf(var[i] * 0.125f + 1e-4f);
}

__global__ __launch_bounds__(256)
void k_bn_apply(float* __restrict__ cbuf, const float* __restrict__ factor,
                const float* __restrict__ bias) {
  size_t e = (size_t)blockIdx.x * 256 + threadIdx.x;
  if (e >= NCPLX_) return;
  int cc = (int)(e & 31);
  size_t r = e >> 5;
  int s = (int)(r & 1);
  size_t rm = r >> 1;
  int m = (int)(rm % 127);
  size_t lb = rm / 127;
  int l = (int)(lb >> 3);
  float f = factor[s * 32 + cc];
  float re = cbuf[e * 2] * f, im = cbuf[e * 2 + 1] * f;
  if (l == 0 && m == 63 && s == 0) re += bias[cc];  // bias row s==0
  cbuf[e * 2] = re;
  cbuf[e * 2 + 1] = im;
}

__global__ __launch_bounds__(256)
void k_add(float* __restrict__ a, const float* __restrict__ b, size_t n) {
  size_t e = (size_t)blockIdx.x * 256 + threadIdx.x;
  if (e < n) a[e] += b[e];
}

// ---------------------------------------------------------------------------
extern "C" void kernel_launch(void* const* d_in, const int* in_sizes, int n_in,
                              void* d_out, int out_size, void* d_ws,
                              size_t ws_size, hipStream_t stream) {
  (void)in_sizes; (void)n_in; (void)out_size; (void)ws_size;
  const float* x       = (const float*)d_in[0];
  const float* leg_fwd = (const float*)d_in[1];
  const float* leg_bwd = (const float*)d_in[2];
  const float* w1_re   = (const float*)d_in[3];
  const float* w1_im   = (const float*)d_in[4];
  const float* w2_re   = (const float*)d_in[5];
  const float* w2_im   = (const float*)d_in[6];
  const float* bn1_s   = (const float*)d_in[7];
  const float* bn1_b   = (const float*)d_in[8];
  const float* bn2_s   = (const float*)d_in[9];
  const float* bn2_b   = (const float*)d_in[10];
  const float* wd      = (const float*)d_in[11];
  const float* bd      = (const float*)d_in[12];
  const float* bnr_s   = (const float*)d_in[13];
  const float* bnr_b   = (const float*)d_in[14];
  float* out = (float*)d_out;  // complex64 interleaved

  // Workspace layout (floats); total ~200 MB.
  float* ws = (float*)d_ws;
  float* wfr = ws;                   // 16384
  float* wfi = wfr + 16384;
  float* wbr = wfi + 16384;
  float* wbi = wbr + 16384;
  float* var = wbi + 16384;          // 64
  float* fac = var + 64;             // 64
  float* fm  = fac + 64;             // 16,646,144  (S*M*N*B*C cplx)
  float* coeffs = fm + 16646144;     // 8,323,072   (residual, kept live)
  float* hA = coeffs + 8323072;      // 8,323,072
  float* hB = hA + 8323072;          // 8,323,072
  float* spat = hB + 8323072;        // 8,388,608 real spatial; reused as res buf
  float* hC = spat;                  // res coefficients (liveness disjoint)

  const size_t smDftF = 36864 * 4;   // 147456 B
  const size_t smLeg  = 8192 * 4;    // 32768 B
  const size_t smMix  = 8192 * 4;    // 32768 B
  const size_t smDftB = 40960 * 4;   // 163840 B  (within 320KB/WG)

  k_init_dft<<<128, 128, 0, stream>>>(wfr, wfi, wbr, wbi);

  // ---- forward 1: coeffs = SWSFT(x) ----
  k_dft_fwd<<<2048, 256, smDftF, stream>>>(x, wfr, wfi, fm);
  k_leg_fwd<<<254, 256, smLeg, stream>>>(leg_fwd, fm, coeffs);

  // ---- h = BN1(coeffs @ w1) ----
  k_chanmix<<<512, 256, smMix, stream>>>(coeffs, w1_re, w1_im, hA);
  k_bn_mean<<<1, 32, 0, stream>>>(hA);
  k_zero_var<<<1, 64, 0, stream>>>(var);
  k_bn_var<<<1016, 256, 0, stream>>>(hA, var);
  k_bn_factor<<<1, 64, 0, stream>>>(var, bn1_s, fac);
  k_bn_apply<<<16256, 256, 0, stream>>>(hA, fac, bn1_b);

  // ---- spatial: collapse(ISWSFT(h)) ----
  k_leg_bwd<<<254, 256, smLeg, stream>>>(leg_bwd, hA, fm);
  k_dft_bwd<<<2048, 256, smDftB, stream>>>(fm, wbr, wbi, spat, nullptr, 0);

  // ---- forward 2 ----
  k_dft_fwd<<<2048, 256, smDftF, stream>>>(spat, wfr, wfi, fm);
  k_leg_fwd<<<254, 256, smLeg, stream>>>(leg_fwd, fm, hA);

  // ---- h = BN2(h @ w2) ----
  k_chanmix<<<512, 256, smMix, stream>>>(hA, w2_re, w2_im, hB);
  k_bn_mean<<<1, 32, 0, stream>>>(hB);
  k_zero_var<<<1, 64, 0, stream>>>(var);
  k_bn_var<<<1016, 256, 0, stream>>>(hB, var);
  k_bn_factor<<<1, 64, 0, stream>>>(var, bn2_s, fac);
  k_bn_apply<<<16256, 256, 0, stream>>>(hB, fac, bn2_b);

  // ---- residual: BNr(coeffs @ wd + bd) ----
  k_resmix<<<1016, 256, 0, stream>>>(coeffs, wd, bd, hC);
  k_bn_mean<<<1, 32, 0, stream>>>(hC);
  k_zero_var<<<1, 64, 0, stream>>>(var);
  k_bn_var<<<1016, 256, 0, stream>>>(hC, var);
  k_bn_factor<<<1, 64, 0, stream>>>(var, bnr_s, fac);
  k_bn_apply<<<16256, 256, 0, stream>>>(hC, fac, bnr_b);

  // ---- out = collapse(ISWSFT(h + res)) ----
  k_add<<<32512, 256, 0, stream>>>(hB, hC, (size_t)NCPLX_ * 2);
  k_leg_bwd<<<254, 256, smLeg, stream>>>(leg_bwd, hB, fm);
  k_dft_bwd<<<2048, 256, smDftB, stream>>>(fm, wbr, wbi, nullptr, out, 1);
}